// HybridQCNN_68135361184093
// MI455X (gfx1250) — compile-verified
//
#include <hip/hip_runtime.h>
#include <math.h>

// MI455X / gfx1250, wave32. Fully fused HybridQCNN forward.
// Memory-bound: ~208 MB traffic -> ~9 us floor @ 23.3 TB/s. All GEMMs run on
// the matrix pipe via V_WMMA_F32_16X16X4_F32 (full fp32, no precision loss).

typedef float v2f __attribute__((ext_vector_type(2)));
typedef float v4f __attribute__((ext_vector_type(4)));
typedef float v8f __attribute__((ext_vector_type(8)));

#define PI_F 3.14159265358979323846f

__global__ __launch_bounds__(256) void hybrid_qcnn_fused(
    const float* __restrict__ x,    // [B,1,28,28]
    const float* __restrict__ w1,   // [32,16]
    const float* __restrict__ b1,   // [32]
    const float* __restrict__ w2,   // [4,32]
    const float* __restrict__ b2,   // [4]
    const float* __restrict__ qw,   // [2,4,3]
    const float* __restrict__ fw1,  // [32,4]
    const float* __restrict__ fb1,  // [32]
    const float* __restrict__ fw2,  // [10,32]
    const float* __restrict__ fb2,  // [10]
    float* __restrict__ out)        // [B,10]
{
    __shared__ float s_h16[256 * 17];  // pooled features (pad 17: bank-conflict free)
    __shared__ float s_buf[256 * 33];  // 32-wide activations (pad 33)
    __shared__ float s_qc[24], s_qs[24];

    const int tid  = threadIdx.x;
    const int lane = tid & 31;
    const int wv   = tid >> 5;
    const int rowb = wv * 32;            // this wave's 32 rows within the block
    const int bb   = blockIdx.x * 256;   // block's batch base
    const int b    = bb + tid;

    const int ln15 = lane & 15;
    const int hi   = lane >> 4;          // 0: lanes 0-15, 1: lanes 16-31
    const int koff = hi * 2;             // A/B K-offset per ISA 16x4 f32 layout

    // ---- uniform trig for circuit weights (once per block) ----
    if (tid < 24) {
        float th = qw[tid] * 0.5f;
        s_qc[tid] = cosf(th);
        s_qs[tid] = sinf(th);
    }

    // ---- stage 1: 28x28 -> 4x4 mean pooling (7x7 bins), aligned float4 loads ----
    {
        const float* xb = x + (size_t)b * 784;
        const float inv49 = 1.0f / 49.0f;
        #pragma unroll
        for (int rb = 0; rb < 4; ++rb) {
            float a0 = 0.f, a1 = 0.f, a2 = 0.f, a3 = 0.f;
            #pragma unroll
            for (int rr = 0; rr < 7; ++rr) {
                const v4f* rp = (const v4f*)(xb + (rb * 7 + rr) * 28);
                v4f v0 = rp[0], v1 = rp[1], v2 = rp[2], v3 = rp[3];
                v4f v4 = rp[4], v5 = rp[5], v6 = rp[6];
                a0 += v0[0] + v0[1] + v0[2] + v0[3] + v1[0] + v1[1] + v1[2];
                a1 += v1[3] + v2[0] + v2[1] + v2[2] + v2[3] + v3[0] + v3[1];
                a2 += v3[2] + v3[3] + v4[0] + v4[1] + v4[2] + v4[3] + v5[0];
                a3 += v5[1] + v5[2] + v5[3] + v6[0] + v6[1] + v6[2] + v6[3];
            }
            s_h16[tid * 17 + rb * 4 + 0] = a0 * inv49;
            s_h16[tid * 17 + rb * 4 + 1] = a1 * inv49;
            s_h16[tid * 17 + rb * 4 + 2] = a2 * inv49;
            s_h16[tid * 17 + rb * 4 + 3] = a3 * inv49;
        }
    }
    __syncthreads();

    // ---- stage 2: h1 = relu(h16 @ w1.T + b1), WMMA f32 16x16x4, K=16 ----
    #pragma unroll
    for (int rt = 0; rt < 2; ++rt) {
        const int arow = rowb + rt * 16 + ln15;   // A-matrix row for this lane
        #pragma unroll
        for (int ct = 0; ct < 2; ++ct) {
            const int n = ct * 16 + ln15;         // output feature (B-matrix col)
            const float bias = b1[n];
            v8f acc = {bias, bias, bias, bias, bias, bias, bias, bias};
            #pragma unroll
            for (int kk = 0; kk < 4; ++kk) {
                const int k0 = kk * 4 + koff;
                v2f A, Bm;
                A[0]  = s_h16[arow * 17 + k0];
                A[1]  = s_h16[arow * 17 + k0 + 1];
                Bm[0] = w1[n * 16 + k0];
                Bm[1] = w1[n * 16 + k0 + 1];
                acc = __builtin_amdgcn_wmma_f32_16x16x4_f32(
                    false, A, false, Bm, (short)0, acc, false, false);
            }
            #pragma unroll
            for (int j = 0; j < 8; ++j) {         // D layout: VGPR j -> M = j (+8 hi)
                const int orow = rowb + rt * 16 + j + hi * 8;
                const float v = acc[j] > 0.f ? acc[j] : 0.f;
                s_buf[orow * 33 + n] = v;
            }
        }
    }
    __syncthreads();

    // ---- stage 3: angles = (tanh(h1 @ w2.T + b2) + 1) * pi/2  (N=4: scalar) ----
    float ang_c[4], ang_s[4];
    #pragma unroll
    for (int q = 0; q < 4; ++q) {
        float acc = b2[q];
        #pragma unroll
        for (int k = 0; k < 32; ++k) acc += s_buf[tid * 33 + k] * w2[q * 32 + k];
        const float half_angle = (tanhf(acc) + 1.0f) * (PI_F * 0.25f);
        ang_c[q] = cosf(half_angle);
        ang_s[q] = sinf(half_angle);
    }

    // ---- stage 4: 4-qubit statevector sim in registers (qubit q = bit 3-q) ----
    float sr[16], si[16];
    #pragma unroll
    for (int i = 0; i < 16; ++i) { sr[i] = (i == 0) ? 1.f : 0.f; si[i] = 0.f; }

    // encoding RY gates
    #pragma unroll
    for (int q = 0; q < 4; ++q) {
        const int bit = 8 >> q;
        const float c = ang_c[q], s = ang_s[q];
        #pragma unroll
        for (int i = 0; i < 16; ++i) {
            if (i & bit) continue;
            const int i1 = i | bit;
            const float r0 = sr[i], r1 = sr[i1], m0 = si[i], m1 = si[i1];
            sr[i]  = c * r0 - s * r1;  si[i]  = c * m0 - s * m1;
            sr[i1] = s * r0 + c * r1;  si[i1] = s * m0 + c * m1;
        }
    }

    #pragma unroll
    for (int l = 0; l < 2; ++l) {
        #pragma unroll
        for (int q = 0; q < 4; ++q) {
            const int bit = 8 >> q;
            {   // RY(qw[l,q,0])
                const float c = s_qc[(l * 4 + q) * 3 + 0], s = s_qs[(l * 4 + q) * 3 + 0];
                #pragma unroll
                for (int i = 0; i < 16; ++i) {
                    if (i & bit) continue;
                    const int i1 = i | bit;
                    const float r0 = sr[i], r1 = sr[i1], m0 = si[i], m1 = si[i1];
                    sr[i]  = c * r0 - s * r1;  si[i]  = c * m0 - s * m1;
                    sr[i1] = s * r0 + c * r1;  si[i1] = s * m0 + c * m1;
                }
            }
            {   // RZ(qw[l,q,1]): diag(e^{-i t/2}, e^{+i t/2})
                const float c = s_qc[(l * 4 + q) * 3 + 1], s = s_qs[(l * 4 + q) * 3 + 1];
                #pragma unroll
                for (int i = 0; i < 16; ++i) {
                    const float r = sr[i], m = si[i];
                    if (i & bit) { sr[i] = c * r - s * m;  si[i] = c * m + s * r; }
                    else         { sr[i] = c * r + s * m;  si[i] = c * m - s * r; }
                }
            }
        }
        // CNOT chain (0,1),(1,2),(2,3)
        #pragma unroll
        for (int q = 0; q < 3; ++q) {
            const int cb = 8 >> q, tb = 8 >> (q + 1);
            #pragma unroll
            for (int i = 0; i < 16; ++i) {
                if ((i & cb) && !(i & tb)) {
                    const int i1 = i | tb;
                    float t;
                    t = sr[i]; sr[i] = sr[i1]; sr[i1] = t;
                    t = si[i]; si[i] = si[i1]; si[i1] = t;
                }
            }
        }
        // RX(qw[l,q,2]): [[c, -i s], [-i s, c]]
        #pragma unroll
        for (int q = 0; q < 4; ++q) {
            const int bit = 8 >> q;
            const float c = s_qc[(l * 4 + q) * 3 + 2], s = s_qs[(l * 4 + q) * 3 + 2];
            #pragma unroll
            for (int i = 0; i < 16; ++i) {
                if (i & bit) continue;
                const int i1 = i | bit;
                const float r0 = sr[i], m0 = si[i], r1 = sr[i1], m1 = si[i1];
                sr[i]  = c * r0 + s * m1;  si[i]  = c * m0 - s * r1;
                sr[i1] = c * r1 + s * m0;  si[i1] = c * m1 - s * r0;
            }
        }
    }

    // <Z_q> readout -> stage into s_h16 (safe: s_h16 reads all completed pre-barrier)
    #pragma unroll
    for (int q = 0; q < 4; ++q) {
        const int bit = 8 >> q;
        float z = 0.f;
        #pragma unroll
        for (int i = 0; i < 16; ++i) {
            const float p = sr[i] * sr[i] + si[i] * si[i];
            z += (i & bit) ? -p : p;
        }
        s_h16[tid * 17 + q] = z;
    }
    __syncthreads();

    // ---- stage 5: z = relu(qf @ fw1.T + fb1), WMMA, K=4 (single chunk) ----
    #pragma unroll
    for (int rt = 0; rt < 2; ++rt) {
        const int arow = rowb + rt * 16 + ln15;
        v2f A;
        A[0] = s_h16[arow * 17 + koff];
        A[1] = s_h16[arow * 17 + koff + 1];
        #pragma unroll
        for (int ct = 0; ct < 2; ++ct) {
            const int n = ct * 16 + ln15;
            const float bias = fb1[n];
            v8f acc = {bias, bias, bias, bias, bias, bias, bias, bias};
            v2f Bm;
            Bm[0] = fw1[n * 4 + koff];
            Bm[1] = fw1[n * 4 + koff + 1];
            acc = __builtin_amdgcn_wmma_f32_16x16x4_f32(
                false, A, false, Bm, (short)0, acc, false, false);
            #pragma unroll
            for (int j = 0; j < 8; ++j) {
                const int orow = rowb + rt * 16 + j + hi * 8;
                const float v = acc[j] > 0.f ? acc[j] : 0.f;
                s_buf[orow * 33 + n] = v;
            }
        }
    }
    __syncthreads();

    // ---- stage 6: logits = z @ fw2.T + fb2, WMMA, K=32, N=10 padded to 16 ----
    {
        const int n = ln15;                       // output class col (0..15, 10 valid)
        const bool nvalid = (n < 10);
        const float bias = nvalid ? fb2[n] : 0.f;
        #pragma unroll
        for (int rt = 0; rt < 2; ++rt) {
            const int arow = rowb + rt * 16 + ln15;
            v8f acc = {bias, bias, bias, bias, bias, bias, bias, bias};
            #pragma unroll
            for (int kk = 0; kk < 8; ++kk) {
                const int k0 = kk * 4 + koff;
                v2f A, Bm;
                A[0]  = s_buf[arow * 33 + k0];
                A[1]  = s_buf[arow * 33 + k0 + 1];
                Bm[0] = nvalid ? fw2[n * 32 + k0]     : 0.f;
                Bm[1] = nvalid ? fw2[n * 32 + k0 + 1] : 0.f;
                acc = __builtin_amdgcn_wmma_f32_16x16x4_f32(
                    false, A, false, Bm, (short)0, acc, false, false);
            }
            #pragma unroll
            for (int j = 0; j < 8; ++j) {
                const int orow = bb + rowb + rt * 16 + j + hi * 8;
                if (nvalid) out[(size_t)orow * 10 + n] = acc[j];
            }
        }
    }
}

extern "C" void kernel_launch(void* const* d_in, const int* in_sizes, int n_in,
                              void* d_out, int out_size, void* d_ws, size_t ws_size,
                              hipStream_t stream) {
    const float* x   = (const float*)d_in[0];
    const float* w1  = (const float*)d_in[1];
    const float* b1  = (const float*)d_in[2];
    const float* w2  = (const float*)d_in[3];
    const float* b2  = (const float*)d_in[4];
    const float* qw  = (const float*)d_in[5];
    const float* fw1 = (const float*)d_in[6];
    const float* fb1 = (const float*)d_in[7];
    const float* fw2 = (const float*)d_in[8];
    const float* fb2 = (const float*)d_in[9];
    float* out = (float*)d_out;

    const int B = in_sizes[0] / 784;   // 65536
    const int blocks = B / 256;        // 256 blocks of 256 threads (8 wave32)
    hybrid_qcnn_fused<<<blocks, 256, 0, stream>>>(
        x, w1, b1, w2, b2, qw, fw1, fb1, fw2, fb2, out);
}